// OrthogonalButterfly_37031208026714
// MI455X (gfx1250) — compile-verified
//
#include <hip/hip_runtime.h>
#include <stdint.h>

// OrthogonalButterfly for MI455X (gfx1250).
// B=32768 rows, WIDTH=1024, HALF=512, DEPTH=20.
// Strategy: each WG pulls 16 rows (64KB) HBM->LDS with CDNA5 async-to-LDS,
// applies all 20 layers in LDS (two layers fused per round trip, ping-pong
// buffers, 10 barriers), then pushes results LDS->HBM with async-from-LDS.
// Memory-bound: 256MB total HBM traffic ~= 11us at 23.3 TB/s.

namespace {
constexpr int WIDTH_   = 1024;
constexpr int HALF_    = 512;
constexpr int DEPTH_   = 20;
constexpr int ROWS_    = 16;    // rows per workgroup (LDS tile)
constexpr int THREADS_ = 256;   // 8 waves (wave32)
constexpr int SWEEPS_  = ROWS_ * WIDTH_ * 4 / (THREADS_ * 16); // 16 b128 sweeps
}

__global__ __launch_bounds__(THREADS_)
void butterfly20_kernel(const float* __restrict__ X,
                        const float* __restrict__ P,
                        float* __restrict__ Y)
{
    // ping-pong row buffers: 2 * 16 * 1024 * 4B = 128 KB LDS
    __shared__ __align__(16) float buf[2][ROWS_][WIDTH_];

    const uint32_t t       = threadIdx.x;
    const uint64_t rowBase = (uint64_t)blockIdx.x * ROWS_;

    // ---- Bulk load: 16 contiguous rows (64KB) -> buf[0] via async-to-LDS ----
    {
        const uint64_t src  = (uint64_t)(uintptr_t)X + rowBase * (WIDTH_ * 4ull);
        // generic LDS address: high dword = shared aperture, low dword = LDS byte offset
        const uint32_t lds0 = (uint32_t)(uintptr_t)(&buf[0][0][0]);
        #pragma unroll
        for (int swp = 0; swp < SWEEPS_; ++swp) {
            uint32_t voff = t * 16u + (uint32_t)swp * (THREADS_ * 16u);
            uint32_t lds  = lds0 + voff;
            // GVS mode: mem = SADDR(u64) + VADDR(i32); LDS dst addr in VGPR
            asm volatile("global_load_async_to_lds_b128 %0, %1, %2"
                         :: "v"(lds), "v"(voff), "s"(src) : "memory");
        }
        asm volatile("s_wait_asynccnt 0x0" ::: "memory");
    }
    __syncthreads();

    // ---- 10 fused layer-pairs ----
    // Even layer d0: thread t owns pairs {t, t+256}; pair i -> cols (2i, 2i+1).
    // Odd layer d1:  thread t then owns pairs {2t, 2t+1} (exactly the values it
    // just produced: a0@2t, a1@2t+1, b0@2t+512, b1@2t+513); outputs are the 4
    // contiguous cols 4t..4t+3 -> single ds_store_b128.
    int srcB = 0;
    #pragma unroll 1
    for (int lp = 0; lp < DEPTH_ / 2; ++lp) {
        const int d0 = 2 * lp;
        const int d1 = d0 + 1;
        // params layout: (HALF, DEPTH) row-major -> P[i*DEPTH + d]
        const float thA = P[(t)           * DEPTH_ + d0];
        const float thB = P[(t + 256u)    * DEPTH_ + d0];
        const float thC = P[(2u * t)      * DEPTH_ + d1];
        const float thD = P[(2u * t + 1u) * DEPTH_ + d1];
        const float cA = __cosf(thA), sA = __sinf(thA);
        const float cB = __cosf(thB), sB = __sinf(thB);
        const float cC = __cosf(thC), sC = __sinf(thC);
        const float cD = __cosf(thD), sD = __sinf(thD);

        const float* __restrict__ bs = &buf[srcB][0][0];
        float*       __restrict__ bd = &buf[srcB ^ 1][0][0];
        #pragma unroll
        for (int r = 0; r < ROWS_; ++r) {
            const float* row = bs + r * WIDTH_;
            const float x0 = row[t];                   // pair t
            const float x1 = row[t + HALF_];
            const float y0 = row[t + 256];             // pair t+256
            const float y1 = row[t + 256 + HALF_];
            // even layer rotations
            const float a0 = x0 * cA + x1 * sA;        // -> col 2t
            const float a1 = x1 * cA - x0 * sA;        // -> col 2t+1
            const float b0 = y0 * cB + y1 * sB;        // -> col 2t+512
            const float b1 = y1 * cB - y0 * sB;        // -> col 2t+513
            // odd layer rotations, outputs cols 4t..4t+3
            float4 o;
            o.x = a0 * cC + b0 * sC;
            o.y = b0 * cC - a0 * sC;
            o.z = a1 * cD + b1 * sD;
            o.w = b1 * cD - a1 * sD;
            *(float4*)(bd + r * WIDTH_ + 4u * t) = o;
        }
        __syncthreads();
        srcB ^= 1;
    }
    // after 10 flips, final data is in buf[0] (srcB == 0)

    // ---- Bulk store: buf[0] -> 16 rows of Y via async-from-LDS ----
    {
        const uint64_t dst  = (uint64_t)(uintptr_t)Y + rowBase * (WIDTH_ * 4ull);
        const uint32_t lds0 = (uint32_t)(uintptr_t)(&buf[0][0][0]);
        #pragma unroll
        for (int swp = 0; swp < SWEEPS_; ++swp) {
            uint32_t voff = t * 16u + (uint32_t)swp * (THREADS_ * 16u);
            uint32_t lds  = lds0 + voff;
            asm volatile("global_store_async_from_lds_b128 %0, %1, %2"
                         :: "v"(voff), "v"(lds), "s"(dst) : "memory");
        }
        asm volatile("s_wait_asynccnt 0x0" ::: "memory");
    }
}

extern "C" void kernel_launch(void* const* d_in, const int* in_sizes, int n_in,
                              void* d_out, int out_size, void* d_ws, size_t ws_size,
                              hipStream_t stream) {
    const float* X = (const float*)d_in[0];   // (32768, 1024) f32
    const float* P = (const float*)d_in[1];   // (512, 20) f32
    float*       Y = (float*)d_out;           // (32768, 1024) f32

    const int nRows = in_sizes[0] / WIDTH_;   // 32768
    const int grid  = nRows / ROWS_;          // 2048 workgroups
    butterfly20_kernel<<<grid, THREADS_, 0, stream>>>(X, P, Y);
}